// StructuralLayer_28484223108091
// MI455X (gfx1250) — compile-verified
//
#include <hip/hip_runtime.h>
#include <stdint.h>

// ---------------------------------------------------------------------------
// Financial scan: B=32768 independent rows, T=60 sequential steps each.
// Memory-bound: ~197 MB compulsory traffic -> ~8.5us floor @ 23.3 TB/s.
// One thread per batch row. Time-chunked (TC=2) double-buffered staging of all
// 17 per-step input arrays through LDS using gfx1250 async tensor-engine loads
// (global_load_async_to_lds_b64, ASYNCcnt-pipelined) so the next chunk's
// memory traffic overlaps the current chunk's recurrence math.
// ---------------------------------------------------------------------------

#define BLOCK   128
#define T_STEPS 60
#define TC      2                     // timesteps per staged chunk (8B/lane/array)
#define NCHUNK  (T_STEPS / TC)        // 30
#define NARR    17                    // per-timestep input arrays

typedef __attribute__((ext_vector_type(4))) float v4f;   // native vector for NT stores

struct Ptrs { const float* p[NARR]; };

__device__ __forceinline__ float relu_f(float x) { return fmaxf(x, 0.0f); }

__global__ __launch_bounds__(BLOCK, 1) void fin_scan_kernel(
    Ptrs ptrs,
    const float* __restrict__ prev_cash, const float* __restrict__ prev_sti,
    const float* __restrict__ prev_st,   const float* __restrict__ prev_lt,
    const float* __restrict__ prev_ar,   const float* __restrict__ prev_ap,
    const float* __restrict__ prev_inv,  const float* __restrict__ prev_ppe,
    const float* __restrict__ prev_eq,   const int* __restrict__ period_days,
    float* __restrict__ out, int Bn)
{
    // 2 buffers x 17 arrays x (128 threads * 2 steps) floats = 34,816 B LDS
    __shared__ float lds[2][NARR][BLOCK * TC];

    const int tid = threadIdx.x;
    const int b   = blockIdx.x * BLOCK + tid;
    if (b >= Bn) return;

    const float inv_pd = 1.0f / (float)(*period_days);

    // Low 32 bits of a flat LDS pointer = workgroup-relative LDS byte offset,
    // which is what the async engine's VDST address operand expects.
    const uint32_t lds_base = (uint32_t)(uintptr_t)(&lds[0][0][0]);

    // Issue 17 async global->LDS b64 loads for chunk c into buffer `buf`.
    auto issue_chunk = [&](int c, int buf) {
        const uint32_t voff  = (uint32_t)((uint32_t)b * T_STEPS + (uint32_t)c * TC) * 4u;
        const uint32_t lbase = lds_base +
            ((uint32_t)buf * NARR * BLOCK * TC + (uint32_t)tid * TC) * 4u;
        #pragma unroll
        for (int a = 0; a < NARR; ++a) {
            const uint32_t laddr = lbase + (uint32_t)(a * BLOCK * TC) * 4u;
            // VDST = per-lane LDS byte address, VADDR = per-lane u32 offset,
            // SADDR = uniform 64-bit array base. Tracked by ASYNCcnt.
            asm volatile("global_load_async_to_lds_b64 %0, %1, %2"
                         :: "v"(laddr), "v"(voff), "s"(ptrs.p[a])
                         : "memory");
        }
    };

    // Carry state (9 balances).
    float cash = prev_cash[b], sti = prev_sti[b], stD = prev_st[b],
          ltD  = prev_lt[b],  ar  = prev_ar[b],  ap  = prev_ap[b],
          inv  = prev_inv[b], ppe = prev_ppe[b], eq  = prev_eq[b];

    issue_chunk(0, 0);                 // prologue: prefetch chunk 0

    for (int c = 0; c < NCHUNK; ++c) {
        const int buf = c & 1;

        if (c + 1 < NCHUNK) {
            // All prior ds reads of buf^1 must land before the async engine
            // overwrites it (async ops are unordered vs DS ops).
            asm volatile("s_wait_dscnt 0x0" ::: "memory");
            issue_chunk(c + 1, buf ^ 1);
            // 17 newer loads may remain outstanding; async loads retire in
            // order, so <=17 means current buffer's 17 loads are complete.
            asm volatile("s_wait_asynccnt 0x11" ::: "memory");
        } else {
            asm volatile("s_wait_asynccnt 0x0" ::: "memory");
        }

        #pragma unroll
        for (int k = 0; k < TC; ++k) {
            const int idx = tid * TC + k;
            const float s   = lds[buf][ 0][idx];
            const float cg  = lds[buf][ 1][idx];
            const float cx  = lds[buf][ 2][idx];
            const float pi  = lds[buf][ 3][idx];
            const float rst = lds[buf][ 4][idx];
            const float rlt = lds[buf][ 5][idx];
            const float tr  = lds[buf][ 6][idx];
            const float mcr = lds[buf][ 7][idx];
            const float lts = lds[buf][ 8][idx];
            const float sis = lds[buf][ 9][idx];
            const float dsp = lds[buf][10][idx];
            const float po  = lds[buf][11][idx];
            const float dso = lds[buf][12][idx];
            const float dpo = lds[buf][13][idx];
            const float dio = lds[buf][14][idx];
            const float orx = lds[buf][15][idx];
            const float dr  = lds[buf][16][idx];

            // Fisher nominal rates
            const float rf_q = (1.0f + relu_f(rst)) * (1.0f + pi) - 1.0f;
            const float sir  = rf_q + sis;
            const float sbr  = rf_q + dsp;
            const float lbr  = (1.0f + relu_f(rlt)) * (1.0f + pi) - 1.0f + dsp;

            // Working-capital targets
            const float ar_e  = relu_f(dso) * inv_pd * relu_f(s);
            const float ap_e  = relu_f(dpo) * inv_pd * relu_f(cg);
            const float inv_e = relu_f(dio) * inv_pd * relu_f(cg);

            // Income statement
            const float opex = orx * s;
            const float dep  = dr * ppe;
            const float ebit = s - cg - opex - dep;
            const float ebt  = ebit + sir * sti - (sbr * stD + lbr * ltD);
            const float tax  = tr * relu_f(ebt);
            const float ni   = ebt - tax;
            const float dv   = po * relu_f(ni);

            // Cash flow + cash/debt plug
            const float ocf      = ni + dep - (ar_e - ar) - (inv_e - inv) + (ap_e - ap);
            const float min_cash = mcr * s;
            const float avail    = cash + sti + ocf - cx - dv;
            const float deficit  = relu_f(min_cash - avail);
            const float surplus  = relu_f(avail - min_cash);
            const float st_e     = stD + (1.0f - lts) * deficit;
            const float lt_e     = ltD + lts * deficit;
            const float cash_e   = fminf(avail + deficit, min_cash);
            const float ppe_e    = ppe - dep + cx;
            const float eq_e     = eq + ni - dv;

            // out[b, t, 0..7] = {ni, cash, sti, st, lt, ppe, eq, ar}; 32B aligned.
            const int t = c * TC + k;
            v4f* op = (v4f*)(out + ((size_t)b * T_STEPS + (size_t)t) * 8);
            v4f o0; o0.x = ni;   o0.y = cash_e; o0.z = surplus; o0.w = st_e;
            v4f o1; o1.x = lt_e; o1.y = ppe_e;  o1.z = eq_e;    o1.w = ar_e;
            __builtin_nontemporal_store(o0, op);
            __builtin_nontemporal_store(o1, op + 1);

            // Advance carry
            cash = cash_e; sti = surplus; stD = st_e; ltD = lt_e;
            ar = ar_e; ap = ap_e; inv = inv_e; ppe = ppe_e; eq = eq_e;
        }
    }
}

extern "C" void kernel_launch(void* const* d_in, const int* in_sizes, int n_in,
                              void* d_out, int out_size, void* d_ws, size_t ws_size,
                              hipStream_t stream) {
    (void)n_in; (void)out_size; (void)d_ws; (void)ws_size;

    Ptrs ptrs;
    for (int i = 0; i < NARR; ++i) ptrs.p[i] = (const float*)d_in[i];

    const float* prev_cash = (const float*)d_in[17];
    const float* prev_sti  = (const float*)d_in[18];
    const float* prev_st   = (const float*)d_in[19];
    const float* prev_lt   = (const float*)d_in[20];
    const float* prev_ar   = (const float*)d_in[21];
    const float* prev_ap   = (const float*)d_in[22];
    const float* prev_inv  = (const float*)d_in[23];
    const float* prev_ppe  = (const float*)d_in[24];
    const float* prev_eq   = (const float*)d_in[25];
    const int*   pdays     = (const int*)d_in[26];

    const int Bn = in_sizes[17];                 // 32768 (prev_cash length)
    const int grid = (Bn + BLOCK - 1) / BLOCK;   // 256 blocks, 1024 waves

    fin_scan_kernel<<<grid, BLOCK, 0, stream>>>(
        ptrs, prev_cash, prev_sti, prev_st, prev_lt, prev_ar, prev_ap,
        prev_inv, prev_ppe, prev_eq, pdays, (float*)d_out, Bn);
}